// Model_1_2_34153579938540
// MI455X (gfx1250) — compile-verified
//
#include <hip/hip_runtime.h>
#include <hip/hip_bf16.h>
#include <math.h>

// ---------------------------------------------------------------------------
// GAT (2 layers) + segment softmax + mean/max graph pooling + 2 MLP heads.
// f32 throughout; dense linears use V_WMMA_F32_16X16X4_F32 (exact f32 WMMA).
// ---------------------------------------------------------------------------

typedef float v2f __attribute__((ext_vector_type(2)));
typedef float v8f __attribute__((ext_vector_type(8)));

#define GCOUNT   128
#define CCH      32
#define DCH      32
#define NEG_ATT  0.2f
#define NEG_ACT  0.01f
#define ORD_NEG_INF 0x007FFFFFu   // f2ord(-inf)

__device__ __forceinline__ unsigned f2ord(float f) {
  unsigned u = __float_as_uint(f);
  return (u & 0x80000000u) ? ~u : (u | 0x80000000u);
}
__device__ __forceinline__ float ord2f(unsigned u) {
  u = (u & 0x80000000u) ? (u & 0x7fffffffu) : ~u;
  return __uint_as_float(u);
}
__device__ __forceinline__ float lrelu(float v, float s) { return v > 0.f ? v : s * v; }

// ---------------------------------------------------------------------------
// Dense linear  out[n][c] = sum_k in[n][k] * W[c][k]   (W is [32, in_ch])
// One wave handles a 16-node tile, both 16-column tiles, via f32 WMMA 16x16x4.
//  A layout (ISA 7.12.2, 32-bit A 16x4): lane L: M=L&15, K=(L>>4)*2 .. +1
//  B layout (K x N, column-striped):     lane L: N=L&15, K=(L>>4)*2 .. +1
//  C/D layout: VGPR i, lane L: row = i + (L>>4)*8, col = L&15
// ---------------------------------------------------------------------------
__global__ void linear_wmma_kernel(const float* __restrict__ in,
                                   const float* __restrict__ W,
                                   float* __restrict__ out,
                                   int n, int in_ch) {
  const int lane = threadIdx.x & 31;
  const int tile = blockIdx.x * (blockDim.x >> 5) + (threadIdx.x >> 5);
  const int m0 = tile * 16;
  if (m0 >= n) return;                       // wave-uniform: EXEC stays all-1s

  const int mrow = lane & 15;
  const int kb   = (lane >> 4) * 2;
  int mload = m0 + mrow;
  if (mload >= n) mload = n - 1;             // clamp for tail tile (stores guarded)

  v8f c0 = {};
  v8f c1 = {};
  const int ksteps = in_ch >> 2;

#if __has_builtin(__builtin_amdgcn_wmma_f32_16x16x4_f32)
  for (int ks = 0; ks < ksteps; ++ks) {
    const int k = ks * 4 + kb;
    v2f a;
    a.x = in[(long)mload * in_ch + k];
    a.y = in[(long)mload * in_ch + k + 1];
    v2f b0, b1;
    b0.x = W[(long)mrow        * in_ch + k];
    b0.y = W[(long)mrow        * in_ch + k + 1];
    b1.x = W[(long)(16 + mrow) * in_ch + k];
    b1.y = W[(long)(16 + mrow) * in_ch + k + 1];
    c0 = __builtin_amdgcn_wmma_f32_16x16x4_f32(false, a, false, b0, (short)0, c0, false, false);
    c1 = __builtin_amdgcn_wmma_f32_16x16x4_f32(false, a, false, b1, (short)0, c1, false, false);
  }
  const int colBase = lane & 15;
  const int rowOff  = (lane >> 4) * 8;
#pragma unroll
  for (int i = 0; i < 8; ++i) {
    int row = m0 + rowOff + i;
    if (row < n) {
      out[(long)row * CCH + colBase]      = c0[i];
      out[(long)row * CCH + 16 + colBase] = c1[i];
    }
  }
#else
  // Scalar fallback (keeps compile alive if builtin absent; histogram flags it)
  for (int r = lane; r < 16; r += 32) {
    int row = m0 + r;
    if (row >= n) break;
    for (int c = 0; c < CCH; ++c) {
      float s = 0.f;
      for (int k = 0; k < in_ch; ++k) s += in[(long)row * in_ch + k] * W[(long)c * in_ch + k];
      out[(long)row * CCH + c] = s;
    }
  }
  (void)c0; (void)c1; (void)ksteps; (void)mload; (void)kb;
#endif
}

// Per-node: attention scalars + init segment-softmax state ------------------
__global__ void node_prep_kernel(const float* __restrict__ h,
                                 const float* __restrict__ a_src,
                                 const float* __restrict__ a_dst,
                                 float* __restrict__ alpha_s,
                                 float* __restrict__ alpha_d,
                                 unsigned* __restrict__ maxb,
                                 float* __restrict__ den, int n) {
  int i = blockIdx.x * blockDim.x + threadIdx.x;
  if (i >= n) return;
  float ss = 0.f, sd = 0.f;
#pragma unroll
  for (int c = 0; c < CCH; ++c) {
    float v = h[(long)i * CCH + c];
    ss += v * a_src[c];
    sd += v * a_dst[c];
  }
  alpha_s[i] = ss;
  alpha_d[i] = sd;
  maxb[i] = ORD_NEG_INF;
  den[i]  = 0.f;
}

__global__ void zero_f32_kernel(float* __restrict__ p, long n) {
  long i = (long)blockIdx.x * blockDim.x + threadIdx.x;
  if (i < n) p[i] = 0.f;
}

// Pass B: edge logits + segment max (ordered-bits atomicMax) ----------------
__global__ void edge_logits_kernel(const int* __restrict__ ei, long E, long EN,
                                   const float* __restrict__ alpha_s,
                                   const float* __restrict__ alpha_d,
                                   float* __restrict__ eedge,
                                   unsigned* __restrict__ maxb) {
  long i = (long)blockIdx.x * blockDim.x + threadIdx.x;
  if (i >= EN) return;
  int src, dst;
  if (i < E) { src = ei[i]; dst = ei[E + i]; }
  else       { src = dst = (int)(i - E); }          // self loop
  float e = lrelu(alpha_s[src] + alpha_d[dst], NEG_ATT);
  eedge[i] = e;
  atomicMax(&maxb[dst], f2ord(e));
}

// Pass C: exp(e - max) + segment sum ----------------------------------------
__global__ void edge_exp_kernel(const int* __restrict__ ei, long E, long EN,
                                float* __restrict__ eedge,
                                const unsigned* __restrict__ maxb,
                                float* __restrict__ den) {
  long i = (long)blockIdx.x * blockDim.x + threadIdx.x;
  if (i >= EN) return;
  int dst = (i < E) ? ei[E + i] : (int)(i - E);
  float ex = expf(eedge[i] - ord2f(maxb[dst]));
  eedge[i] = ex;
  atomicAdd(&den[dst], ex);
}

// Pass D: out[dst] += h[src] * alpha   (one wave per edge, lane = channel) --
__global__ void edge_aggregate_kernel(const int* __restrict__ ei, long E, long EN,
                                      const float* __restrict__ eedge,
                                      const float* __restrict__ den,
                                      const float* __restrict__ h,
                                      float* __restrict__ out) {
  long t = (long)blockIdx.x * blockDim.x + threadIdx.x;
  long e = t >> 5;
  int lane = (int)(t & 31);
  if (e >= EN) return;
  int src, dst;
  if (e < E) { src = ei[e]; dst = ei[E + e]; }
  else       { src = dst = (int)(e - E); }
  float alpha = eedge[e] / (den[dst] + 1e-16f);
  atomicAdd(&out[(long)dst * CCH + lane], h[(long)src * CCH + lane] * alpha);
}

// Bias + LeakyReLU(0.01) ----------------------------------------------------
__global__ void bias_act_kernel(const float* __restrict__ acc,
                                const float* __restrict__ bias,
                                float* __restrict__ outp, long total) {
  long i = (long)blockIdx.x * blockDim.x + threadIdx.x;
  if (i >= total) return;
  float v = acc[i] + bias[i & (CCH - 1)];
  outp[i] = lrelu(v, NEG_ACT);
}

// Pooling -------------------------------------------------------------------
__global__ void pool_init_kernel(float* __restrict__ meansum,
                                 unsigned* __restrict__ maxpool,
                                 float* __restrict__ cnt) {
  int i = blockIdx.x * blockDim.x + threadIdx.x;
  if (i < GCOUNT * CCH) { meansum[i] = 0.f; maxpool[i] = ORD_NEG_INF; }
  if (i < GCOUNT) cnt[i] = 0.f;
}

#define POOL_CHUNK 512
__global__ void pool_kernel(const float* __restrict__ act,
                            const int* __restrict__ batch,
                            float* __restrict__ meansum,
                            unsigned* __restrict__ maxpool,
                            float* __restrict__ cnt, int n) {
  int lane = threadIdx.x & 31;
  long wave = ((long)blockIdx.x * blockDim.x + threadIdx.x) >> 5;
  long start = wave * POOL_CHUNK;
  if (start >= n) return;
  long end = start + POOL_CHUNK; if (end > n) end = n;

  int curg = -1, count = 0;
  float sum = 0.f, mx = -INFINITY;
  for (long i = start; i < end; ++i) {
    int g = batch[i];                        // wave-uniform (sorted batching)
    if (g != curg) {
      if (curg >= 0) {
        atomicAdd(&meansum[curg * CCH + lane], sum);
        atomicMax(&maxpool[curg * CCH + lane], f2ord(mx));
        if (lane == 0) atomicAdd(&cnt[curg], (float)count);
      }
      curg = g; sum = 0.f; mx = -INFINITY; count = 0;
    }
    float v = act[i * CCH + lane];
    sum += v;
    mx = fmaxf(mx, v);
    ++count;
  }
  if (curg >= 0) {
    atomicAdd(&meansum[curg * CCH + lane], sum);
    atomicMax(&maxpool[curg * CCH + lane], f2ord(mx));
    if (lane == 0) atomicAdd(&cnt[curg], (float)count);
  }
}

// Heads: g = [mean | max] (G x 64) -> two 3-layer MLPs -> out [G, 2] --------
__global__ void head_kernel(const float* __restrict__ meansum,
                            const unsigned* __restrict__ maxpool,
                            const float* __restrict__ cnt,
                            const float* __restrict__ Wc1, const float* __restrict__ bc1,
                            const float* __restrict__ Wc2, const float* __restrict__ bc2,
                            const float* __restrict__ Wc3, const float* __restrict__ bc3,
                            const float* __restrict__ Wr1, const float* __restrict__ br1,
                            const float* __restrict__ Wr2, const float* __restrict__ br2,
                            const float* __restrict__ Wr3, const float* __restrict__ br3,
                            float* __restrict__ out) {
  int g = threadIdx.x;
  if (g >= GCOUNT) return;
  float gv[2 * CCH];
  float inv = 1.f / fmaxf(cnt[g], 1.f);
  for (int c = 0; c < CCH; ++c) {
    gv[c]       = meansum[g * CCH + c] * inv;
    gv[CCH + c] = ord2f(maxpool[g * CCH + c]);
  }
  float t1[DCH], t2[DCH];
  // chi head
  for (int d = 0; d < DCH; ++d) {
    float s = bc1[d];
    for (int k = 0; k < 2 * CCH; ++k) s += gv[k] * Wc1[d * 2 * CCH + k];
    t1[d] = lrelu(s, NEG_ACT);
  }
  for (int d = 0; d < DCH; ++d) {
    float s = bc2[d];
    for (int k = 0; k < DCH; ++k) s += t1[k] * Wc2[d * DCH + k];
    t2[d] = lrelu(s, NEG_ACT);
  }
  float chi = bc3[0];
  for (int k = 0; k < DCH; ++k) chi += t2[k] * Wc3[k];
  // rp head
  for (int d = 0; d < DCH; ++d) {
    float s = br1[d];
    for (int k = 0; k < 2 * CCH; ++k) s += gv[k] * Wr1[d * 2 * CCH + k];
    t1[d] = lrelu(s, NEG_ACT);
  }
  for (int d = 0; d < DCH; ++d) {
    float s = br2[d];
    for (int k = 0; k < DCH; ++k) s += t1[k] * Wr2[d * DCH + k];
    t2[d] = lrelu(s, NEG_ACT);
  }
  float rp = br3[0];
  for (int k = 0; k < DCH; ++k) rp += t2[k] * Wr3[k];

  out[g * 2 + 0] = chi;
  out[g * 2 + 1] = rp;
}

// ---------------------------------------------------------------------------
extern "C" void kernel_launch(void* const* d_in, const int* in_sizes, int n_in,
                              void* d_out, int out_size, void* d_ws, size_t ws_size,
                              hipStream_t stream) {
  const float* X    = (const float*)d_in[0];
  const int*   EI   = (const int*)  d_in[1];
  // d_in[2] = Edge_weight: unused (GATConv without edge_dim)
  const int*   Bat  = (const int*)  d_in[3];
  const float* W1   = (const float*)d_in[4];
  const float* as1  = (const float*)d_in[5];
  const float* ad1  = (const float*)d_in[6];
  const float* b1   = (const float*)d_in[7];
  const float* W2   = (const float*)d_in[8];
  const float* as2  = (const float*)d_in[9];
  const float* ad2  = (const float*)d_in[10];
  const float* b2   = (const float*)d_in[11];
  const float* Wc1  = (const float*)d_in[12];
  const float* bc1  = (const float*)d_in[13];
  const float* Wc2  = (const float*)d_in[14];
  const float* bc2  = (const float*)d_in[15];
  const float* Wc3  = (const float*)d_in[16];
  const float* bc3  = (const float*)d_in[17];
  const float* Wr1  = (const float*)d_in[18];
  const float* br1  = (const float*)d_in[19];
  const float* Wr2  = (const float*)d_in[20];
  const float* br2  = (const float*)d_in[21];
  const float* Wr3  = (const float*)d_in[22];
  const float* br3  = (const float*)d_in[23];

  const int  N  = in_sizes[0] / 4;
  const long E  = in_sizes[2];
  const long EN = E + (long)N;
  const long NC = (long)N * CCH;

  // Workspace carve-up (floats / u32, all 4-byte)
  float*    ws       = (float*)d_ws;
  float*    bufH     = ws;                 // [N, 32] linear output (both layers)
  float*    bufO     = bufH + NC;          // [N, 32] aggregation accumulator
  float*    bufA     = bufO + NC;          // [N, 32] activated layer output
  float*    alpha_s  = bufA + NC;          // [N]
  float*    alpha_d  = alpha_s + N;        // [N]
  float*    den      = alpha_d + N;        // [N]
  unsigned* maxb     = (unsigned*)(den + N);          // [N]
  float*    eedge    = (float*)(maxb + N);            // [E+N]
  float*    meansum  = eedge + EN;                    // [G, 32]
  unsigned* maxpool  = (unsigned*)(meansum + GCOUNT * CCH);  // [G, 32]
  float*    cnt      = (float*)(maxpool + GCOUNT * CCH);     // [G]

  const int tiles      = (N + 15) / 16;
  const int linBlocks  = (tiles + 3) / 4;            // 4 waves (128 thr) / block
  const int nodeBlocks = (N + 255) / 256;
  const int ncBlocks   = (int)((NC + 255) / 256);
  const int edgeBlocks = (int)((EN + 255) / 256);
  const int aggBlocks  = (int)((EN * 32 + 255) / 256);
  const long poolWaves = ((long)N + POOL_CHUNK - 1) / POOL_CHUNK;
  const int poolBlocks = (int)((poolWaves + 7) / 8);

  // ---------------- Layer 1 ----------------
  linear_wmma_kernel<<<linBlocks, 128, 0, stream>>>(X, W1, bufH, N, 4);
  node_prep_kernel<<<nodeBlocks, 256, 0, stream>>>(bufH, as1, ad1, alpha_s, alpha_d, maxb, den, N);
  zero_f32_kernel<<<ncBlocks, 256, 0, stream>>>(bufO, NC);
  edge_logits_kernel<<<edgeBlocks, 256, 0, stream>>>(EI, E, EN, alpha_s, alpha_d, eedge, maxb);
  edge_exp_kernel<<<edgeBlocks, 256, 0, stream>>>(EI, E, EN, eedge, maxb, den);
  edge_aggregate_kernel<<<aggBlocks, 256, 0, stream>>>(EI, E, EN, eedge, den, bufH, bufO);
  bias_act_kernel<<<ncBlocks, 256, 0, stream>>>(bufO, b1, bufA, NC);

  // ---------------- Layer 2 ----------------
  linear_wmma_kernel<<<linBlocks, 128, 0, stream>>>(bufA, W2, bufH, N, 32);
  node_prep_kernel<<<nodeBlocks, 256, 0, stream>>>(bufH, as2, ad2, alpha_s, alpha_d, maxb, den, N);
  zero_f32_kernel<<<ncBlocks, 256, 0, stream>>>(bufO, NC);
  edge_logits_kernel<<<edgeBlocks, 256, 0, stream>>>(EI, E, EN, alpha_s, alpha_d, eedge, maxb);
  edge_exp_kernel<<<edgeBlocks, 256, 0, stream>>>(EI, E, EN, eedge, maxb, den);
  edge_aggregate_kernel<<<aggBlocks, 256, 0, stream>>>(EI, E, EN, eedge, den, bufH, bufO);
  bias_act_kernel<<<ncBlocks, 256, 0, stream>>>(bufO, b2, bufA, NC);

  // ---------------- Pooling + heads ----------------
  pool_init_kernel<<<(GCOUNT * CCH + 255) / 256, 256, 0, stream>>>(meansum, maxpool, cnt);
  pool_kernel<<<poolBlocks, 256, 0, stream>>>(bufA, Bat, meansum, maxpool, cnt, N);
  head_kernel<<<1, 128, 0, stream>>>(meansum, maxpool, cnt,
                                     Wc1, bc1, Wc2, bc2, Wc3, bc3,
                                     Wr1, br1, Wr2, br2, Wr3, br3,
                                     (float*)d_out);
  (void)n_in; (void)out_size; (void)ws_size;
}